// TestModel_73349451481893
// MI455X (gfx1250) — compile-verified
//
#include <hip/hip_runtime.h>
#include <math.h>

typedef __attribute__((ext_vector_type(2))) float v2f;
typedef __attribute__((ext_vector_type(8))) float v8f;

namespace {
constexpr int kB   = 512;
constexpr int kNB  = 4;
constexpr int kNQ  = 12;
constexpr int kDIM = 4096;
// state output: (B, NB, 4, DIM, 1) complex64 -> 2 floats each
constexpr size_t kStateFloats = (size_t)2 * kB * kNB * 4 * kDIM;  // 67,108,864
}

// One workgroup per (b, n). Computes the product-state expansion, streams the
// full 4x4096 complex state tile to HBM (coalesced b128 stores), and computes
// G[b,n,:,:] = S * W via one v_wmma_f32_16x16x4_f32 in wave 0.
__global__ __launch_bounds__(256) void qsim_state_kernel(const float* __restrict__ x,
                                                         float* __restrict__ out,
                                                         float* __restrict__ gws) {
  const int bn = blockIdx.x;      // 0 .. B*NB-1
  const int n  = bn & 3;
  const int t  = threadIdx.x;     // 0 .. 255

  __shared__ float vc[12];
  __shared__ float vs[12];
  __shared__ float w0c_s, w0s_s, w1c_s, w1s_s;

  // Stage 1: per-qubit composed rotation vectors.
  // qubit q maps to bit (11-q) of the basis index d.
  if (t < 12) {
    const int b = bn >> 2;
    const float x0 = x[((size_t)b * kNB + n) * kNQ + t];          // layer 0
    const float x1 = x[((size_t)(kB + b) * kNB + n) * kNQ + t];   // layer 1
    float s0, c0, s1, c1;
    sincosf(0.5f * x0, &s0, &c0);
    sincosf(0.5f * x1, &s1, &c1);
    vc[t] = c1 * c0 - s1 * s0;   // R(x1) * u(x0), component 0
    vs[t] = s1 * c0 + c1 * s0;   // component 1
    if (t == 0) {
      // qubit 0 (d-bit 11) channel vectors:
      w0c_s = c1 * c0 - s1 * s0;  // channel c=0: R*u
      w0s_s = s1 * c0 + c1 * s0;
      w1c_s = c1 * c0 + s1 * s0;  // channel c=1: R*(E*u), E=diag(1,-1)
      w1s_s = s1 * c0 - c1 * s0;
    }
  }
  __syncthreads();

  const float w0c = w0c_s, w0s = w0s_s, w1c = w1c_s, w1s = w1s_s;

  // Channel top-qubit vectors: k = c + 2*m; entangle sign only for n < 2.
  float wkc[4], wks[4];
  if (n < 2) {
    wkc[0] = w0c; wks[0] =  w0s;
    wkc[1] = w1c; wks[1] =  w1s;
    wkc[2] = w0c; wks[2] = -w0s;
    wkc[3] = w1c; wks[3] = -w1s;
  } else {
#pragma unroll
    for (int k = 0; k < 4; ++k) { wkc[k] = w0c; wks[k] = w0s; }
  }

  // Stage 2: tensor-product expansion.
  // d' (low 11 bits of d): bit0 <- j, bits1..8 <- t, bits9..10 <- s
  // d-bit j corresponds to qubit (11 - j).
  float Pf = 1.0f;
#pragma unroll
  for (int bt = 0; bt < 8; ++bt) {     // d'-bits 1..8 -> qubits 10..3
    Pf *= ((t >> bt) & 1) ? vs[10 - bt] : vc[10 - bt];
  }
  const float tj0 = Pf * vc[11];       // d'-bit 0 -> qubit 11
  const float tj1 = Pf * vs[11];
  float Sf[4];
#pragma unroll
  for (int s = 0; s < 4; ++s) {        // d'-bits 9,10 -> qubits 2,1
    Sf[s] = ((s & 1) ? vs[2] : vc[2]) * ((s & 2) ? vs[1] : vc[1]);
  }

  // Stage 3: fully-coalesced state writes (real, imag=0).
  float* sbase = out + (size_t)bn * 4 * kDIM * 2;
#pragma unroll
  for (int k = 0; k < 4; ++k) {
#pragma unroll
    for (int h = 0; h < 2; ++h) {      // h = d-bit 11 (qubit 0)
      const float sc = h ? wks[k] : wkc[k];
      const float a0 = sc * tj0;
      const float a1 = sc * tj1;
#pragma unroll
      for (int s = 0; s < 4; ++s) {
        float4 f;
        f.x = a0 * Sf[s];  f.y = 0.0f;
        f.z = a1 * Sf[s];  f.w = 0.0f;
        // lane t writes 16 contiguous bytes; wave covers 512 contiguous bytes
        *(float4*)(sbase + ((size_t)k * kDIM + h * 2048 + s * 512 + 2 * t) * 2) = f;
      }
    }
  }

  // Stage 4: G = S * W via WMMA (wave 0 only; uniform branch keeps EXEC full).
  if (t < 32) {
    float S = 1.0f;
#pragma unroll
    for (int q = 1; q < 12; ++q) S *= (vc[q] * vc[q] - vs[q] * vs[q]);

    const int lane = t;
    const int kk = lane & 3;
    // per-lane channel vector for row/col index kk
    float wc_k, ws_k;
    if (n < 2) {
      wc_k = (kk & 1) ? w1c : w0c;
      float tmp = (kk & 1) ? w1s : w0s;
      ws_k = (kk & 2) ? -tmp : tmp;
    } else {
      wc_k = w0c; ws_k = w0s;
    }

    // A (16x4): rows k<4 = [wk0, wk1, wk1, 0]  (documented 16x4 f32 layout)
    // B packed so both plausible 4x16 layouts decode to rows {wl0, -wl1} in
    // the K-columns that A pairs with -> C[k][l] = wk0*wl0 - wk1*wl1 = W[k][l]
    v2f a;  a.x = 0.0f;  a.y = 0.0f;
    v2f bb; bb.x = 0.0f; bb.y = 0.0f;
    if (lane < 4) {
      a.x = wc_k;  a.y = ws_k;          // A[k][0], A[k][1]
      bb.x = wc_k; bb.y = -ws_k;        // B row 0 / row 1(or 2)
    } else if (lane >= 16 && lane < 20) {
      a.x = ws_k;                       // A[k][2] (lanes 16-31 hold K=2)
    }

    v8f c = {0.f, 0.f, 0.f, 0.f, 0.f, 0.f, 0.f, 0.f};
    c = __builtin_amdgcn_wmma_f32_16x16x4_f32(
        /*neg_a=*/false, a, /*neg_b=*/false, bb,
        /*c_mod=*/(short)0, c, /*reuse_a=*/false, /*reuse_b=*/false);

    // C/D layout: VGPR r, lanes 0-15 -> element [M=r][N=lane]
    if (lane < 4) {
      float* g = gws + (size_t)bn * 16;
      g[0 * 4 + lane] = S * c[0];
      g[1 * 4 + lane] = S * c[1];
      g[2 * 4 + lane] = S * c[2];
      g[3 * 4 + lane] = S * c[3];
    }
  }
}

// One wave per batch b: P = prod_n G[b,n], O[b] = sum_{k,l} P[k,l]*conj(a_k)*a_l.
// conj(ALPHA)⊗ALPHA = kron(M,M), M = [[1,-i],[i,1]]/2 (phases of a0 cancel).
__global__ __launch_bounds__(32) void qsim_reduce_kernel(const float* __restrict__ gws,
                                                         float* __restrict__ out) {
  const int b = blockIdx.x;
  const int lane = threadIdx.x;

  float ore = 0.0f, oim = 0.0f;
  if (lane < 16) {
    float p = gws[((size_t)b * 4 + 0) * 16 + lane];
    p *= gws[((size_t)b * 4 + 1) * 16 + lane];
    p *= gws[((size_t)b * 4 + 2) * 16 + lane];
    p *= gws[((size_t)b * 4 + 3) * 16 + lane];

    const int k = lane >> 2, l = lane & 3;
    const int k1 = k >> 1, k0 = k & 1, l1 = l >> 1, l0 = l & 1;
    // m(a,b): diag -> 0.5 ; (0,1) -> -0.5i ; (1,0) -> +0.5i
    const float r1 = (k1 == l1) ? 0.5f : 0.0f;
    const float i1 = (k1 == l1) ? 0.0f : ((k1 < l1) ? -0.5f : 0.5f);
    const float r0 = (k0 == l0) ? 0.5f : 0.0f;
    const float i0 = (k0 == l0) ? 0.0f : ((k0 < l0) ? -0.5f : 0.5f);
    const float m2re = r1 * r0 - i1 * i0;
    const float m2im = r1 * i0 + i1 * r0;
    ore = p * m2re;
    oim = p * m2im;
  }
#pragma unroll
  for (int off = 8; off; off >>= 1) {
    ore += __shfl_xor(ore, off, 32);
    oim += __shfl_xor(oim, off, 32);
  }
  if (lane == 0) {
    out[kStateFloats + 2 * (size_t)b]     = ore;
    out[kStateFloats + 2 * (size_t)b + 1] = oim;
  }
}

extern "C" void kernel_launch(void* const* d_in, const int* in_sizes, int n_in,
                              void* d_out, int out_size, void* d_ws, size_t ws_size,
                              hipStream_t stream) {
  const float* x = (const float*)d_in[0];   // (2, B, NB, NQ) float32
  float* out = (float*)d_out;               // state floats then O floats
  float* gws = (float*)d_ws;                // B*NB*16 floats = 128 KB scratch

  qsim_state_kernel<<<kB * kNB, 256, 0, stream>>>(x, out, gws);
  qsim_reduce_kernel<<<kB, 32, 0, stream>>>(gws, out);
}